// COS_Loss_45423574122758
// MI455X (gfx1250) — compile-verified
//
#include <hip/hip_runtime.h>
#include <hip/hip_bf16.h>
#include <math.h>

typedef __attribute__((ext_vector_type(2))) float v2f;
typedef __attribute__((ext_vector_type(4))) float v4f;
typedef __attribute__((ext_vector_type(8))) float v8f;

#define KWIN     128
#define WSTART   768          // (nwr-1)*K = 6*128 : start of the last window
#define WW       1024
#define HH       1024
#define NPLANES  24           // B*C = 8*3
#define CHUNKS   8            // row-chunks per plane -> 192 blocks
#define NBLOCKS  (NPLANES * CHUNKS)
#define ROWS_PER_CHUNK (KWIN / CHUNKS)   // 16 rows
#define NWAVES   8            // 256 threads / wave32
#define COUNT_F  49.0f        // nwr*nwc = 7*7

// Exact full-wave (32-lane) sum of v via V_WMMA_F32_16X16X4_F32.
// A = {v,0} per lane (any bijective A layout works: only sum(A) is used).
// B = all-ones (layout-invariant). With ones-B, D[m][n] = rowsum_m(A); per the
// documented C/D layout, summing a lane's 8 D registers yields the M=0..7
// total on lanes 0-15 and the M=8..15 total on lanes 16-31; one xor-16
// shuffle + add gives the exact 32-lane sum on every lane.
__device__ __forceinline__ float wave_sum_wmma(float v) {
    v2f a; a[0] = v;    a[1] = 0.0f;
    v2f b; b[0] = 1.0f; b[1] = 1.0f;
    v8f c = {};
    c = __builtin_amdgcn_wmma_f32_16x16x4_f32(
            /*neg_a=*/false, a, /*neg_b=*/false, b,
            /*c_mod=*/(short)0, c, /*reuse_a=*/false, /*reuse_b=*/false);
    float half = c[0] + c[1] + c[2] + c[3] + c[4] + c[5] + c[6] + c[7];
    float other = __shfl_xor(half, 16, 32);
    return half + other;
}

// 192 blocks: blockIdx -> (plane, 16-row chunk) of the last 128x128 window.
// Each block reduces 16 rows x 128 cols of x and t, writes 3 floats to ws.
__global__ void cosloss_window_partials(const float* __restrict__ x,
                                        const float* __restrict__ t,
                                        float* __restrict__ ws) {
    const int plane = blockIdx.x >> 3;                  // 0..23
    const int chunk = blockIdx.x & 7;                   // 0..7
    const size_t base = (size_t)plane * HH * WW
                      + (size_t)(WSTART + chunk * ROWS_PER_CHUNK) * WW
                      + WSTART;

    float dot = 0.0f, ni = 0.0f, nt = 0.0f;

    // 16 rows x 32 float4 per row = 512 vec4 / 256 threads = 2 iters.
    // 32 consecutive lanes cover one 512B row segment (fully coalesced).
    const int nvec = ROWS_PER_CHUNK * (KWIN / 4);
    for (int v = threadIdx.x; v < nvec; v += blockDim.x) {
        const int wi = (v & 31) << 2;                   // 0..124 step 4
        const int hi = v >> 5;                          // 0..15
        const size_t off = base + (size_t)hi * WW + wi;
        v4f xv = *(const v4f*)(x + off);
        v4f tv = *(const v4f*)(t + off);
#pragma unroll
        for (int k = 0; k < 4; ++k) {
            dot = fmaf(xv[k], tv[k], dot);
            ni  = fmaf(xv[k], xv[k], ni);
            nt  = fmaf(tv[k], tv[k], nt);
        }
    }

    // Uniform control flow so far -> EXEC all ones for WMMA.
    const float wdot = wave_sum_wmma(dot);
    const float wni  = wave_sum_wmma(ni);
    const float wnt  = wave_sum_wmma(nt);

    __shared__ float s_part[NWAVES][3];
    if ((threadIdx.x & 31) == 0) {
        const int wave = threadIdx.x >> 5;
        s_part[wave][0] = wdot;
        s_part[wave][1] = wni;
        s_part[wave][2] = wnt;
    }
    __syncthreads();

    if (threadIdx.x == 0) {
        float bd = 0.0f, bi = 0.0f, bt = 0.0f;
#pragma unroll
        for (int wv = 0; wv < NWAVES; ++wv) {           // fixed order
            bd += s_part[wv][0];
            bi += s_part[wv][1];
            bt += s_part[wv][2];
        }
        float* slot = ws + (size_t)blockIdx.x * 4;      // 16B slots
        slot[0] = bd;
        slot[1] = bi;
        slot[2] = bt;
    }
}

// One wave32: deterministic strided sum of the 192 block partials, then the
// same WMMA wave reduction; lane 0 computes the final scalar.
__global__ void cosloss_finalize(const float* __restrict__ ws,
                                 float* __restrict__ out) {
    float dot = 0.0f, ni = 0.0f, nt = 0.0f;
    for (int i = threadIdx.x; i < NBLOCKS; i += 32) {   // 6 slots per lane
        dot += ws[i * 4 + 0];
        ni  += ws[i * 4 + 1];
        nt  += ws[i * 4 + 2];
    }
    const float td = wave_sum_wmma(dot);
    const float ti = wave_sum_wmma(ni);
    const float tt = wave_sum_wmma(nt);
    if (threadIdx.x == 0) {
        const float cosv = td / (sqrtf(ti) * sqrtf(tt));
        const float d = cosv - 1.0f;
        out[0] = d * d / COUNT_F;
    }
}

extern "C" void kernel_launch(void* const* d_in, const int* in_sizes, int n_in,
                              void* d_out, int out_size, void* d_ws, size_t ws_size,
                              hipStream_t stream) {
    (void)in_sizes; (void)n_in; (void)out_size; (void)ws_size;
    const float* x = (const float*)d_in[0];   // input  (8,3,1024,1024) f32
    const float* t = (const float*)d_in[1];   // target (8,3,1024,1024) f32
    float* ws  = (float*)d_ws;                // NBLOCKS*4 floats = 3 KB
    float* out = (float*)d_out;               // 1 float

    cosloss_window_partials<<<NBLOCKS, 256, 0, stream>>>(x, t, ws);
    cosloss_finalize<<<1, 32, 0, stream>>>(ws, out);
}